// PtModule_76166950028001
// MI455X (gfx1250) — compile-verified
//
#include <hip/hip_runtime.h>
#include <hip/hip_bf16.h>

// Elementwise: out = min(((x + 1) * 0.75)^2, 10.0f)
// 8192x8192 fp32 -> pure HBM-streaming kernel (~512 MB traffic, ~22us floor at 23.3 TB/s).
//
// Main kernel: branch-free full tiles -> clean global_load_b128/store_b128 NT
// stream with no exec-mask churn. Tail kernel (guarded, scalar) only launches
// when n isn't tile-divisible (never for the reference shape).

typedef float v4f __attribute__((ext_vector_type(4)));

#define UNROLL 4
#define BLOCK 256
#define TILE_V4 (BLOCK * UNROLL)        // v4f elements per block
#define TILE_F  (TILE_V4 * 4)           // float elements per block

__device__ __forceinline__ float ew_op(float x) {
    float t = fmaf(0.75f, x, 0.75f);   // (x + 1) * 0.75
    float y = t * t;                   // squared
    return fminf(y, 10.0f);            // clamp above at 10
}

// Full tiles only: no bounds checks anywhere.
__global__ __launch_bounds__(BLOCK) void ew_main_kernel(
    const v4f* __restrict__ x, v4f* __restrict__ out) {

    int base = (int)(blockIdx.x) * TILE_V4 + (int)threadIdx.x;

    v4f v[UNROLL];

    // 4 outstanding non-temporal global_load_b128 per thread.
#pragma unroll
    for (int k = 0; k < UNROLL; ++k) {
        v[k] = __builtin_nontemporal_load(x + base + k * BLOCK);
    }

#pragma unroll
    for (int k = 0; k < UNROLL; ++k) {
        v4f r;
        r.x = ew_op(v[k].x);
        r.y = ew_op(v[k].y);
        r.z = ew_op(v[k].z);
        r.w = ew_op(v[k].w);
        __builtin_nontemporal_store(r, out + base + k * BLOCK);
    }
}

// Scalar guarded tail for the (rare) non-divisible remainder.
__global__ __launch_bounds__(BLOCK) void ew_tail_kernel(
    const float* __restrict__ x, float* __restrict__ out, int start, int n) {

    int i = start + (int)blockIdx.x * BLOCK + (int)threadIdx.x;
    if (i < n) {
        out[i] = ew_op(x[i]);
    }
}

extern "C" void kernel_launch(void* const* d_in, const int* in_sizes, int n_in,
                              void* d_out, int out_size, void* d_ws, size_t ws_size,
                              hipStream_t stream) {
    (void)n_in; (void)d_ws; (void)ws_size; (void)out_size;

    const float* x = (const float*)d_in[0];
    float* out = (float*)d_out;

    int n = in_sizes[0];               // 8192*8192 = 67,108,864 (divisible by TILE_F)
    int full_tiles = n / TILE_F;       // 16384 for the reference shape
    int covered = full_tiles * TILE_F;

    if (full_tiles > 0) {
        ew_main_kernel<<<full_tiles, BLOCK, 0, stream>>>(
            (const v4f*)x, (v4f*)out);
    }
    int rem = n - covered;
    if (rem > 0) {
        int tail_blocks = (rem + BLOCK - 1) / BLOCK;
        ew_tail_kernel<<<tail_blocks, BLOCK, 0, stream>>>(x, out, covered, n);
    }
}